// DynamicGraphLearner_6219112644629
// MI455X (gfx1250) — compile-verified
//
#include <hip/hip_runtime.h>
#include <math.h>

#define NUM_NODES  2048
#define NODE_DIM   64
#define BATCH      16
#define TROWS      16
#define LDS_STRIDE 2052   // 2048 + 4 pad: rows 8 apart differ by 32 banks; 2052*4 % 16 == 0

typedef __attribute__((ext_vector_type(2))) float v2f;
typedef __attribute__((ext_vector_type(4))) float v4f;
typedef __attribute__((ext_vector_type(8))) float v8f;

// ---------------------------------------------------------------------------
// Prep: node_1 = emb@w1^T + b1, node_2 = emb@w2^T + b2, v = x@wp^T + bp
// Tiny (300K dot-64s) -> plain VALU.
// ---------------------------------------------------------------------------
__global__ __launch_bounds__(256) void dgl_prep(
    const float* __restrict__ x_temp, const float* __restrict__ node_emb,
    const float* __restrict__ w1_w, const float* __restrict__ w1_b,
    const float* __restrict__ w2_w, const float* __restrict__ w2_b,
    const float* __restrict__ wp_w, const float* __restrict__ wp_b,
    float* __restrict__ n1, float* __restrict__ n2, float* __restrict__ v)
{
    const int tid = blockIdx.x * blockDim.x + threadIdx.x;
    const int NE  = NUM_NODES * NODE_DIM;
    if (tid < NE) {
        const int i = tid >> 6, j = tid & 63;
        float acc = 0.0f;
        #pragma unroll 8
        for (int k = 0; k < NODE_DIM; ++k)
            acc += node_emb[i * NODE_DIM + k] * w1_w[j * NODE_DIM + k];
        n1[tid] = acc + w1_b[j];
    } else if (tid < 2 * NE) {
        const int t2 = tid - NE;
        const int i = t2 >> 6, j = t2 & 63;
        float acc = 0.0f;
        #pragma unroll 8
        for (int k = 0; k < NODE_DIM; ++k)
            acc += node_emb[i * NODE_DIM + k] * w2_w[j * NODE_DIM + k];
        n2[t2] = acc + w2_b[j];
    } else if (tid < 2 * NE + BATCH * NUM_NODES) {
        const int t3 = tid - 2 * NE;
        float acc = 0.0f;
        #pragma unroll 8
        for (int k = 0; k < NODE_DIM; ++k)
            acc += x_temp[(size_t)t3 * NODE_DIM + k] * wp_w[k];
        v[t3] = acc + wp_b[0];
    }
}

// ---------------------------------------------------------------------------
// adj_base = relu(n1 @ n2^T) : 2048x2048, computed ONCE (batch-invariant).
// One wave per 16x16 tile; WMMA f32 16x16x4, K=64 -> 16 steps.
// 16 MB result stays L2-resident for the 16 batch passes of the main kernel.
// ---------------------------------------------------------------------------
__global__ __launch_bounds__(256) void dgl_abase(
    const float* __restrict__ n1, const float* __restrict__ n2,
    float* __restrict__ abase)
{
    const int tid  = threadIdx.x;
    const int wave = tid >> 5;
    const int lane = tid & 31;
    const int lo   = lane & 15;
    const int hi   = lane >> 4;

    const int rblk  = blockIdx.x >> 4;                 // 0..127
    const int cgrp  = blockIdx.x & 15;                 // 0..15
    const int rbase = rblk * 16;
    const int cbase = (cgrp * 8 + wave) * 16;

    v2f a[16], bf[16];
    {
        const float* pa = n1 + (size_t)(rbase + lo) * NODE_DIM + 2 * hi;
        const float* pb = n2 + (size_t)(cbase + lo) * NODE_DIM + 2 * hi;
        #pragma unroll
        for (int s = 0; s < 16; ++s) {
            a[s]  = *(const v2f*)(pa + 4 * s);
            bf[s] = *(const v2f*)(pb + 4 * s);
        }
    }
    v8f acc = {0.f, 0.f, 0.f, 0.f, 0.f, 0.f, 0.f, 0.f};
    #pragma unroll
    for (int s = 0; s < 16; ++s)
        acc = __builtin_amdgcn_wmma_f32_16x16x4_f32(
            false, a[s], false, bf[s], (short)0, acc, false, false);

    #pragma unroll
    for (int i = 0; i < 8; ++i) {
        const int r = i + 8 * hi;
        const float vx = acc[i] > 0.0f ? acc[i] : 0.0f;
        abase[(size_t)(rbase + r) * NUM_NODES + cbase + lo] = vx;
    }
}

// ---------------------------------------------------------------------------
// Fused: adj = adj_base + x@x^T + (v_i - v_j); softmax over columns.
// One workgroup = one (batch, 16-row slab). 8 waves, each owns 256 columns.
// USE_ABASE=1: read precomputed relu(n1 n2^T) tile (L2-resident).
// USE_ABASE=0: recompute it with a second WMMA chain (workspace fallback).
// Slab kept in LDS; softmax in LDS; nontemporal f32x4 writeback (protect L2).
// ---------------------------------------------------------------------------
template <bool USE_ABASE>
__global__ __launch_bounds__(256) void dgl_adj_softmax(
    const float* __restrict__ x_temp,
    const float* __restrict__ n1,
    const float* __restrict__ n2,
    const float* __restrict__ abase,
    const float* __restrict__ v,
    float* __restrict__ out)
{
    extern __shared__ float smem[];
    float* tileBuf = smem;                          // TROWS * LDS_STRIDE
    float* part    = smem + TROWS * LDS_STRIDE;     // 16 x 16 partials
    float* rowstat = part + 256;                    // 16 (max, then 1/sum)

    const int b     = blockIdx.y;
    const int rbase = blockIdx.x * TROWS;
    const int tid   = threadIdx.x;
    const int wave  = tid >> 5;
    const int lane  = tid & 31;
    const int lo    = lane & 15;   // row/col within 16
    const int hi    = lane >> 4;   // half-wave: K sub-slot / M offset of 8

    const float* xb = x_temp + (size_t)b * NUM_NODES * NODE_DIM;
    const float* vb = v + b * NUM_NODES;

    // A fragments (rows of this slab), resident across the whole column loop.
    // f32 16x4 A layout: lane(lo) = M, VGPR pair = {K=4s+2*hi, K=4s+2*hi+1}.
    v2f ax[16], an[16];
    {
        const float* pax = xb + (size_t)(rbase + lo) * NODE_DIM + 2 * hi;
        #pragma unroll
        for (int s = 0; s < 16; ++s) ax[s] = *(const v2f*)(pax + 4 * s);
        if constexpr (!USE_ABASE) {
            const float* pan = n1 + (size_t)(rbase + lo) * NODE_DIM + 2 * hi;
            #pragma unroll
            for (int s = 0; s < 16; ++s) an[s] = *(const v2f*)(pan + 4 * s);
        }
    }

    // v for this lane's 8 accumulator rows (C layout: VGPR i -> M = i + 8*hi)
    float vrow[8];
    #pragma unroll
    for (int i = 0; i < 8; ++i) vrow[i] = vb[rbase + i + 8 * hi];

    for (int t = 0; t < 16; ++t) {
        const int cbase = (wave * 16 + t) * 16;
        const float* pbx = xb + (size_t)(cbase + lo) * NODE_DIM + 2 * hi;
        if (t + 1 < 16)  // global_prefetch_b8 of next tile's B operand
            __builtin_prefetch(pbx + 16 * NODE_DIM, 0, 1);

        // Issue ALL fragment loads before the WMMA chain (max loads in flight).
        v2f bx[16];
        #pragma unroll
        for (int s = 0; s < 16; ++s) bx[s] = *(const v2f*)(pbx + 4 * s);

        float basev[8];
        v2f bn[16];
        if constexpr (USE_ABASE) {
            const float* pb0 =
                abase + (size_t)(rbase + 8 * hi) * NUM_NODES + cbase + lo;
            #pragma unroll
            for (int i = 0; i < 8; ++i) basev[i] = pb0[(size_t)i * NUM_NODES];
        } else {
            const float* pbn = n2 + (size_t)(cbase + lo) * NODE_DIM + 2 * hi;
            #pragma unroll
            for (int s = 0; s < 16; ++s) bn[s] = *(const v2f*)(pbn + 4 * s);
        }

        v8f accd = {0.f, 0.f, 0.f, 0.f, 0.f, 0.f, 0.f, 0.f};  // dyn = x x^T
        #pragma unroll
        for (int s = 0; s < 16; ++s)
            accd = __builtin_amdgcn_wmma_f32_16x16x4_f32(
                false, ax[s], false, bx[s], (short)0, accd, false, false);

        v8f accb;
        if constexpr (!USE_ABASE) {
            accb = (v8f){0.f, 0.f, 0.f, 0.f, 0.f, 0.f, 0.f, 0.f};
            #pragma unroll
            for (int s = 0; s < 16; ++s)
                accb = __builtin_amdgcn_wmma_f32_16x16x4_f32(
                    false, an[s], false, bn[s], (short)0, accb, false, false);
        }

        const float vcol = vb[cbase + lo];
        #pragma unroll
        for (int i = 0; i < 8; ++i) {
            const int r = i + 8 * hi;
            float base;
            if constexpr (USE_ABASE)
                base = basev[i];
            else
                base = accb[i] > 0.0f ? accb[i] : 0.0f;   // relu BEFORE add
            tileBuf[r * LDS_STRIDE + cbase + lo] = base + accd[i] + vrow[i] - vcol;
        }
    }
    __syncthreads();

    // ---- softmax over the 16 x 2048 slab in LDS ----
    const int r = tid >> 4;   // row 0..15
    const int j = tid & 15;   // 16 reducers per row

    float m = -3.0e38f;
    for (int c = j; c < NUM_NODES; c += 16)
        m = fmaxf(m, tileBuf[r * LDS_STRIDE + c]);
    part[r * 16 + j] = m;
    __syncthreads();
    if (j == 0) {
        float mm = part[r * 16];
        for (int k = 1; k < 16; ++k) mm = fmaxf(mm, part[r * 16 + k]);
        rowstat[r] = mm;
    }
    __syncthreads();

    const float rmax = rowstat[r];
    float s = 0.0f;
    for (int c = j; c < NUM_NODES; c += 16) {
        const float e = __expf(tileBuf[r * LDS_STRIDE + c] - rmax);
        tileBuf[r * LDS_STRIDE + c] = e;
        s += e;
    }
    part[r * 16 + j] = s;
    __syncthreads();
    if (j == 0) {
        float ss = 0.0f;
        for (int k = 0; k < 16; ++k) ss += part[r * 16 + k];
        rowstat[r] = 1.0f / ss;
    }
    __syncthreads();

    // ---- normalized writeback: streaming 256 MB -> nontemporal stores so the
    // ---- L2-resident adj_base / x_temp working set is not evicted.
    const size_t obase = ((size_t)b * NUM_NODES + rbase) * NUM_NODES;
    for (int row = 0; row < TROWS; ++row) {
        const float sc = rowstat[row];
        const v4f* src = (const v4f*)(tileBuf + row * LDS_STRIDE);
        v4f* dst = (v4f*)(out + obase + (size_t)row * NUM_NODES);
        for (int c = tid; c < NUM_NODES / 4; c += 256) {
            v4f val = src[c] * sc;
            __builtin_nontemporal_store(val, dst + c);
        }
    }
}

extern "C" void kernel_launch(void* const* d_in, const int* in_sizes, int n_in,
                              void* d_out, int out_size, void* d_ws, size_t ws_size,
                              hipStream_t stream) {
    (void)in_sizes; (void)n_in; (void)out_size;

    const float* x_temp   = (const float*)d_in[0];
    const float* node_emb = (const float*)d_in[1];
    const float* w1_w     = (const float*)d_in[2];
    const float* w1_b     = (const float*)d_in[3];
    const float* w2_w     = (const float*)d_in[4];
    const float* w2_b     = (const float*)d_in[5];
    const float* wp_w     = (const float*)d_in[6];
    const float* wp_b     = (const float*)d_in[7];
    float* out = (float*)d_out;

    float* n1 = (float*)d_ws;                     // 2048*64
    float* n2 = n1 + NUM_NODES * NODE_DIM;        // 2048*64
    float* v  = n2 + NUM_NODES * NODE_DIM;        // 16*2048
    float* ab = v  + BATCH * NUM_NODES;           // 2048*2048 (optional)

    const size_t baseWords  = 2 * NUM_NODES * NODE_DIM + BATCH * NUM_NODES;
    const size_t needAbase  = (baseWords + (size_t)NUM_NODES * NUM_NODES) * sizeof(float);
    const bool   useAbase   = (ws_size >= needAbase);   // ws_size is fixed -> deterministic

    const int totalPrep = 2 * NUM_NODES * NODE_DIM + BATCH * NUM_NODES;
    dgl_prep<<<(totalPrep + 255) / 256, 256, 0, stream>>>(
        x_temp, node_emb, w1_w, w1_b, w2_w, w2_b, wp_w, wp_b, n1, n2, v);

    const size_t smemBytes = (size_t)(TROWS * LDS_STRIDE + 256 + 16) * sizeof(float);
    dim3 grid(NUM_NODES / TROWS, BATCH);

    if (useAbase) {
        dgl_abase<<<(NUM_NODES / 16) * 16, 256, 0, stream>>>(n1, n2, ab);
        (void)hipFuncSetAttribute(reinterpret_cast<const void*>(dgl_adj_softmax<true>),
                                  hipFuncAttributeMaxDynamicSharedMemorySize, (int)smemBytes);
        dgl_adj_softmax<true><<<grid, 256, smemBytes, stream>>>(
            x_temp, n1, n2, ab, v, out);
    } else {
        (void)hipFuncSetAttribute(reinterpret_cast<const void*>(dgl_adj_softmax<false>),
                                  hipFuncAttributeMaxDynamicSharedMemorySize, (int)smemBytes);
        dgl_adj_softmax<false><<<grid, 256, smemBytes, stream>>>(
            x_temp, n1, n2, n1 /*unused*/, v, out);
    }
}